// CrossAttentionPoseRegression_21268678050292
// MI455X (gfx1250) — compile-verified
//
#include <hip/hip_runtime.h>
#include <hip/hip_bf16.h>

#define NPTS 8192
#define FDIM 35
#define F2   70
#define MPAD 80
#define C1   2048
#define C2   128
#define NCOR 2048

typedef float v2f __attribute__((ext_vector_type(2)));
typedef float v8f __attribute__((ext_vector_type(8)));

// ---------------- workspace layout (float offsets) ----------------
#define WS_A    0                           // 80*8192 padded transposed features
#define WS_U    (WS_A + MPAD*NPTS)          // 70*2048 gemm1 out / relu
#define WS_V    (WS_U + F2*C1)              // 70*128  gemm2 out (cs/ct transposed)
#define WS_CSN  (WS_V + F2*C2)              // 128*35  normalized cs rows
#define WS_CTN  (WS_CSN + C2*FDIM)          // 128*35
#define WS_SIM  (WS_CTN + C2*FDIM)          // 128*128
#define WS_FEAT (WS_SIM + C2*C2)            // 8960
#define WS_Z1   (WS_FEAT + C2*F2)           // 256 gemv accumulators
#define WS_CORR (WS_Z1 + 256)               // scalar
#define WS_RANK (WS_CORR + 1)               // scalar

// ---------------- init: zero accumulators every call ----------------
__global__ void k_init(float* __restrict__ ws) {
  int idx = blockIdx.x * 256 + threadIdx.x;
  if (idx < F2 * C1) ws[WS_U + idx] = 0.0f;
  if (idx < 258) {
    if (idx < 256) ws[WS_Z1 + idx] = 0.0f;
    else if (idx == 256) ws[WS_CORR] = 0.0f;
    else ws[WS_RANK] = 0.0f;
  }
}

// ---------------- build A = [hs^T ; ht^T] padded to 80 rows ----------------
__global__ void k_build_A(const float* __restrict__ hs, const float* __restrict__ xs,
                          const float* __restrict__ ht, const float* __restrict__ xt,
                          float* __restrict__ A) {
  int n = blockIdx.x * 256 + threadIdx.x;
  int f = blockIdx.y;
  float v = 0.0f;
  if (f < 32)      v = hs[n * 32 + f];
  else if (f < 35) v = xs[n * 3 + (f - 32)];
  else if (f < 67) v = ht[n * 32 + (f - 35)];
  else if (f < 70) v = xt[n * 3 + (f - 67)];
  A[f * NPTS + n] = v;
}

// ---------------- GEMM1: U(70x2048) += A(80x8192) @ Wc1(8192x2048), f32 WMMA, split-K ----------------
__global__ void __launch_bounds__(256) k_gemm1(const float* __restrict__ A,
                                               const float* __restrict__ W,
                                               float* __restrict__ U) {
  const int wave = threadIdx.x >> 5;
  const int lane = threadIdx.x & 31;
  const int lo = lane & 15;
  const int hi = lane >> 4;
  const int n  = blockIdx.x * 128 + wave * 16 + lo;   // output column
  const int k0 = blockIdx.y * 1024;                   // K split
  v8f acc[5] = {};
  for (int k = k0; k < k0 + 1024; k += 4) {
    const int kb = k + 2 * hi;
    v2f b;
    b[0] = W[(size_t)kb * C1 + n];
    b[1] = W[(size_t)(kb + 1) * C1 + n];
#pragma unroll
    for (int mt = 0; mt < 5; ++mt) {
      const float* ap = A + (mt * 16 + lo) * NPTS + kb;
      v2f a; a[0] = ap[0]; a[1] = ap[1];
      acc[mt] = __builtin_amdgcn_wmma_f32_16x16x4_f32(
          false, a, false, b, (short)0, acc[mt], false, false);
    }
  }
#pragma unroll
  for (int mt = 0; mt < 5; ++mt) {
#pragma unroll
    for (int r = 0; r < 8; ++r) {
      int m = mt * 16 + r + 8 * hi;
      if (m < F2) unsafeAtomicAdd(&U[m * C1 + n], acc[mt][r]);
    }
  }
}

// ---------------- bias + relu on U (in place) ----------------
__global__ void k_bias_relu(float* __restrict__ U, const float* __restrict__ bc1) {
  int idx = blockIdx.x * 256 + threadIdx.x;   // < 70*2048
  int c = idx % C1;
  U[idx] = fmaxf(U[idx] + bc1[c], 0.0f);
}

// ---------------- GEMM2: V(70x128) = U(70x2048) @ Wc2(2048x128) + bc2 ----------------
__global__ void k_gemm2(const float* __restrict__ U, const float* __restrict__ Wc2,
                        const float* __restrict__ bc2, float* __restrict__ V) {
  int idx = blockIdx.x * 256 + threadIdx.x;   // < 8960
  int f = idx / C2, p = idx % C2;
  const float* u = U + f * C1;
  float s = bc2[p];
  for (int k = 0; k < C1; ++k) s = fmaf(u[k], Wc2[k * C2 + p], s);
  V[idx] = s;
}

// ---------------- normalize cs/ct rows and small sim = cs_n @ ct_n^T ----------------
__global__ void __launch_bounds__(256) k_norm_sim(const float* __restrict__ V,
                                                  float* __restrict__ csn,
                                                  float* __restrict__ ctn,
                                                  float* __restrict__ sim) {
  __shared__ float ns[C2], nt[C2];
  int t = threadIdx.x;
  if (t < C2) {
    float s = 0.0f;
    for (int f = 0; f < FDIM; ++f) { float v = V[f * C2 + t]; s = fmaf(v, v, s); }
    ns[t] = fmaxf(sqrtf(s), 1e-12f);
  } else {
    int p = t - C2;
    float s = 0.0f;
    for (int f = FDIM; f < F2; ++f) { float v = V[f * C2 + p]; s = fmaf(v, v, s); }
    nt[p] = fmaxf(sqrtf(s), 1e-12f);
  }
  __syncthreads();
  for (int idx = t; idx < C2 * FDIM; idx += 256) {
    int p = idx / FDIM, f = idx % FDIM;
    csn[idx] = V[f * C2 + p] / ns[p];
    ctn[idx] = V[(FDIM + f) * C2 + p] / nt[p];
  }
  __syncthreads();
  for (int idx = t; idx < C2 * C2; idx += 256) {
    int i = idx >> 7, j = idx & (C2 - 1);
    float s = 0.0f;
    for (int f = 0; f < FDIM; ++f) s = fmaf(csn[i * FDIM + f], ctn[j * FDIM + f], s);
    sim[idx] = s;
  }
}

// ---------------- corr loss: only the 2048 gathered entries of sim_big ----------------
__global__ void __launch_bounds__(256) k_corr(const float* __restrict__ hs, const float* __restrict__ xs,
                                              const float* __restrict__ ht, const float* __restrict__ xt,
                                              const int* __restrict__ corr, const float* __restrict__ labels,
                                              float* __restrict__ corrSum) {
  __shared__ float red[256];
  int t = threadIdx.x;
  int idx = blockIdx.x * 256 + t;           // < 2048
  int i = corr[2 * idx], j = corr[2 * idx + 1];
  float dot = 0.0f, n1 = 0.0f, n2 = 0.0f;
  for (int f = 0; f < 32; ++f) {
    float a = hs[i * 32 + f], b = ht[j * 32 + f];
    dot = fmaf(a, b, dot); n1 = fmaf(a, a, n1); n2 = fmaf(b, b, n2);
  }
  for (int f = 0; f < 3; ++f) {
    float a = xs[i * 3 + f], b = xt[j * 3 + f];
    dot = fmaf(a, b, dot); n1 = fmaf(a, a, n1); n2 = fmaf(b, b, n2);
  }
  float sv = dot / (fmaxf(sqrtf(n1), 1e-12f) * fmaxf(sqrtf(n2), 1e-12f));
  float e = sv - labels[idx];
  red[t] = e * e;
  __syncthreads();
  for (int s = 128; s > 0; s >>= 1) {
    if (t < s) red[t] += red[t + s];
    __syncthreads();
  }
  if (t == 0) unsafeAtomicAdd(corrSum, red[0]);
}

// ---------------- rank loss via 35x35 eigenproblem (sim has rank <= 35) ----------------
__global__ void __launch_bounds__(256) k_svd(const float* __restrict__ csn,
                                             const float* __restrict__ ctn,
                                             float* __restrict__ rankOut) {
  __shared__ float Pm[FDIM * FDIM], Qm[FDIM * FDIM], Lm[FDIM * FDIM], Tm[FDIM * FDIM];
  __shared__ float S[36 * 36];
  __shared__ float cA[18], sA[18];
  __shared__ int ppA[18], qqA[18];
  int t = threadIdx.x;
  for (int idx = t; idx < 36 * 36; idx += 256) S[idx] = 0.0f;
  for (int idx = t; idx < FDIM * FDIM; idx += 256) Lm[idx] = 0.0f;
  __syncthreads();
  // P = Cs^T Cs, Q = Ct^T Ct  (Cs = csn 128x35 row-major)
  for (int idx = t; idx < 2 * FDIM * FDIM; idx += 256) {
    int m = idx % (FDIM * FDIM);
    int a = m / FDIM, b = m % FDIM;
    const float* src = (idx < FDIM * FDIM) ? csn : ctn;
    float s = 0.0f;
    for (int p = 0; p < C2; ++p) s = fmaf(src[p * FDIM + a], src[p * FDIM + b], s);
    if (idx < FDIM * FDIM) Pm[m] = s; else Qm[m] = s;
  }
  __syncthreads();
  // Cholesky Q = L L^T
  for (int j = 0; j < FDIM; ++j) {
    if (t == 0) {
      float d = Qm[j * FDIM + j] + 1e-8f;
      for (int k = 0; k < j; ++k) d -= Lm[j * FDIM + k] * Lm[j * FDIM + k];
      Lm[j * FDIM + j] = sqrtf(fmaxf(d, 1e-20f));
    }
    __syncthreads();
    int i = j + 1 + t;
    if (i < FDIM) {
      float s = Qm[i * FDIM + j];
      for (int k = 0; k < j; ++k) s -= Lm[i * FDIM + k] * Lm[j * FDIM + k];
      Lm[i * FDIM + j] = s / Lm[j * FDIM + j];
    }
    __syncthreads();
  }
  // T = P L ; S = L^T T  (eig(S) = nonzero eig(P Q) = sigma(sim)^2)
  for (int idx = t; idx < FDIM * FDIM; idx += 256) {
    int i = idx / FDIM, j = idx % FDIM;
    float s = 0.0f;
    for (int b = j; b < FDIM; ++b) s = fmaf(Pm[i * FDIM + b], Lm[b * FDIM + j], s);
    Tm[idx] = s;
  }
  __syncthreads();
  for (int idx = t; idx < FDIM * FDIM; idx += 256) {
    int i = idx / FDIM, j = idx % FDIM;
    float s = 0.0f;
    for (int a = i; a < FDIM; ++a) s = fmaf(Lm[a * FDIM + i], Tm[a * FDIM + j], s);
    S[i * 36 + j] = s;
  }
  __syncthreads();
  // parallel cyclic Jacobi on 36x36 (index 35 is a zero pad), 12 sweeps
  for (int sweep = 0; sweep < 12; ++sweep) {
    for (int r = 0; r < FDIM; ++r) {
      if (t < 18) {
        int p_, q_;
        if (t == 0) { p_ = r; q_ = FDIM; }
        else {
          int a = (r + t) % FDIM, b = (r + FDIM - t) % FDIM;
          p_ = a < b ? a : b; q_ = a < b ? b : a;
        }
        ppA[t] = p_; qqA[t] = q_;
        float apq = S[p_ * 36 + q_];
        float c = 1.0f, s = 0.0f;
        if (fabsf(apq) > 1e-30f) {
          float tau = (S[q_ * 36 + q_] - S[p_ * 36 + p_]) / (2.0f * apq);
          float tt = ((tau >= 0.0f) ? 1.0f : -1.0f) / (fabsf(tau) + sqrtf(1.0f + tau * tau));
          c = 1.0f / sqrtf(1.0f + tt * tt);
          s = tt * c;
        }
        cA[t] = c; sA[t] = s;
      }
      __syncthreads();
      for (int idx = t; idx < 18 * 36; idx += 256) {     // S <- S*J (disjoint column pairs)
        int pr = idx / 36, row = idx % 36;
        int p_ = ppA[pr], q_ = qqA[pr];
        float c = cA[pr], s = sA[pr];
        float xp = S[row * 36 + p_], xq = S[row * 36 + q_];
        S[row * 36 + p_] = c * xp - s * xq;
        S[row * 36 + q_] = s * xp + c * xq;
      }
      __syncthreads();
      for (int idx = t; idx < 18 * 36; idx += 256) {     // S <- J^T*S (disjoint row pairs)
        int pr = idx / 36, col = idx % 36;
        int p_ = ppA[pr], q_ = qqA[pr];
        float c = cA[pr], s = sA[pr];
        float xp = S[p_ * 36 + col], xq = S[q_ * 36 + col];
        S[p_ * 36 + col] = c * xp - s * xq;
        S[q_ * 36 + col] = s * xp + c * xq;
      }
      __syncthreads();
    }
  }
  if (t == 0) {
    float acc = 0.0f;
    for (int i = 0; i < FDIM; ++i) {
      float sv = sqrtf(fmaxf(S[i * 36 + i], 0.0f));
      float d = sv - 1.0f;
      acc += d * d;
    }
    acc += (float)(C2 - FDIM);      // 93 zero singular values: (0-1)^2 each
    rankOut[0] = acc / (float)C2;
  }
}

// ---------------- feat = concat(sim^T @ cs, sim @ ct).reshape(-1) ----------------
__global__ void k_feat(const float* __restrict__ sim, const float* __restrict__ V,
                       float* __restrict__ feat) {
  int idx = blockIdx.x * 256 + threadIdx.x;   // < 8960
  int p = idx / F2, f = idx % F2;
  float s = 0.0f;
  const float* vrow = V + f * C2;             // cs[k][f] or ct[k][f-35] == V[f][k]
  if (f < FDIM) { for (int k = 0; k < C2; ++k) s = fmaf(sim[k * C2 + p], vrow[k], s); }
  else          { for (int k = 0; k < C2; ++k) s = fmaf(sim[p * C2 + k], vrow[k], s); }
  feat[idx] = s;
}

// ---------------- GEMV: z1 += feat @ Wp1 (split-K over 35 blocks) ----------------
__global__ void __launch_bounds__(256) k_gemv1(const float* __restrict__ feat,
                                               const float* __restrict__ Wp1,
                                               float* __restrict__ z1) {
  __shared__ float fs[256];
  int k0 = blockIdx.x * 256;
  int j = threadIdx.x;
  fs[j] = feat[k0 + j];
  __syncthreads();
  float s = 0.0f;
  for (int kk = 0; kk < 256; ++kk) s = fmaf(fs[kk], Wp1[(k0 + kk) * 256 + j], s);
  unsafeAtomicAdd(&z1[j], s);
}

// ---------------- MLP tail + loss assembly ----------------
__global__ void __launch_bounds__(256) k_tail(const float* __restrict__ z1acc, const float* __restrict__ bp1,
                                              const float* __restrict__ Wp2, const float* __restrict__ bp2,
                                              const float* __restrict__ Wp3, const float* __restrict__ bp3,
                                              const float* __restrict__ Wp4, const float* __restrict__ bp4,
                                              const float* __restrict__ corrSum, const float* __restrict__ rank,
                                              float* __restrict__ out) {
  __shared__ float a1[256], a2[128], a3[64];
  int t = threadIdx.x;
  a1[t] = fmaxf(z1acc[t] + bp1[t], 0.0f);
  __syncthreads();
  if (t < 128) {
    float s = bp2[t];
    for (int k = 0; k < 256; ++k) s = fmaf(a1[k], Wp2[k * 128 + t], s);
    a2[t] = fmaxf(s, 0.0f);
  }
  __syncthreads();
  if (t < 64) {
    float s = bp3[t];
    for (int k = 0; k < 128; ++k) s = fmaf(a2[k], Wp3[k * 64 + t], s);
    a3[t] = fmaxf(s, 0.0f);
  }
  __syncthreads();
  if (t < 7) {
    float s = bp4[t];
    for (int k = 0; k < 64; ++k) s = fmaf(a3[k], Wp4[k * 7 + t], s);
    out[t] = s;                      // quaternion[0:4], translation[4:7]
  }
  if (t == 7) out[7] = corrSum[0] * (1.0f / (float)NCOR) + rank[0];
}

// ---------------- launcher ----------------
extern "C" void kernel_launch(void* const* d_in, const int* in_sizes, int n_in,
                              void* d_out, int out_size, void* d_ws, size_t ws_size,
                              hipStream_t stream) {
  const float* h_src = (const float*)d_in[0];
  const float* x_src = (const float*)d_in[1];
  const float* h_tgt = (const float*)d_in[2];
  const float* x_tgt = (const float*)d_in[3];
  const int*   corr  = (const int*)d_in[4];
  const float* labels= (const float*)d_in[5];
  const float* Wc1 = (const float*)d_in[6];
  const float* bc1 = (const float*)d_in[7];
  const float* Wc2 = (const float*)d_in[8];
  const float* bc2 = (const float*)d_in[9];
  const float* Wp1 = (const float*)d_in[10];
  const float* bp1 = (const float*)d_in[11];
  const float* Wp2 = (const float*)d_in[12];
  const float* bp2 = (const float*)d_in[13];
  const float* Wp3 = (const float*)d_in[14];
  const float* bp3 = (const float*)d_in[15];
  const float* Wp4 = (const float*)d_in[16];
  const float* bp4 = (const float*)d_in[17];
  float* ws  = (float*)d_ws;
  float* out = (float*)d_out;

  k_init<<<562, 256, 0, stream>>>(ws);
  k_build_A<<<dim3(NPTS / 256, MPAD), 256, 0, stream>>>(h_src, x_src, h_tgt, x_tgt, ws + WS_A);
  k_gemm1<<<dim3(16, 8), 256, 0, stream>>>(ws + WS_A, Wc1, ws + WS_U);
  k_bias_relu<<<(F2 * C1) / 256, 256, 0, stream>>>(ws + WS_U, bc1);
  k_gemm2<<<(F2 * C2) / 256, 256, 0, stream>>>(ws + WS_U, Wc2, bc2, ws + WS_V);
  k_norm_sim<<<1, 256, 0, stream>>>(ws + WS_V, ws + WS_CSN, ws + WS_CTN, ws + WS_SIM);
  k_corr<<<NCOR / 256, 256, 0, stream>>>(h_src, x_src, h_tgt, x_tgt, corr, labels, ws + WS_CORR);
  k_svd<<<1, 256, 0, stream>>>(ws + WS_CSN, ws + WS_CTN, ws + WS_RANK);
  k_feat<<<(C2 * F2) / 256, 256, 0, stream>>>(ws + WS_SIM, ws + WS_V, ws + WS_FEAT);
  k_gemv1<<<(C2 * F2) / 256, 256, 0, stream>>>(ws + WS_FEAT, Wp1, ws + WS_Z1);
  k_tail<<<1, 256, 0, stream>>>(ws + WS_Z1, bp1, Wp2, bp2, Wp3, bp3, Wp4, bp4,
                                ws + WS_CORR, ws + WS_RANK, out);
}